// GeneratorLoss_49898930045229
// MI455X (gfx1250) — compile-verified
//
#include <hip/hip_runtime.h>
#include <hip/hip_bf16.h>

// out[i] = -log(1 - (x[i]-1)^2), fp32 elementwise, memory-bound streaming kernel.
// MI455X: wave32, 128-bit NT global loads/stores, hardware v_log_f32.

#define NEG_LN2 (-0.69314718055994530942f)

// clang-native vector type: required by __builtin_nontemporal_load/store
typedef __attribute__((ext_vector_type(4))) float v4f;

__device__ __forceinline__ float neg_log_one_minus_sq(float v) {
    float d = v - 1.0f;                     // x - s  (s == 1.0)
    float t = __builtin_fmaf(-d, d, 1.0f);  // 1 - d*d   (single FMA)
    // v_log_f32 computes log2; -ln(t) = -ln2 * log2(t)
    return NEG_LN2 * __builtin_amdgcn_logf(t);
}

__global__ __launch_bounds__(256) void GeneratorLoss_elemwise_kernel(
    const float* __restrict__ x, float* __restrict__ out, int n4, int n) {
    const int tid    = blockIdx.x * blockDim.x + threadIdx.x;
    const int stride = gridDim.x * blockDim.x;

    const v4f* __restrict__ x4 = reinterpret_cast<const v4f*>(x);
    v4f* __restrict__       o4 = reinterpret_cast<v4f*>(out);

    // Vectorized body: 128-bit non-temporal load -> 4x (sub, fma, v_log_f32, mul)
    // -> 128-bit non-temporal store. Data is touched once; NT hint avoids
    // polluting the 192MB L2 with a 268MB stream.
    for (int i = tid; i < n4; i += stride) {
        v4f v = __builtin_nontemporal_load(&x4[i]);
        v4f r;
        r.x = neg_log_one_minus_sq(v.x);
        r.y = neg_log_one_minus_sq(v.y);
        r.z = neg_log_one_minus_sq(v.z);
        r.w = neg_log_one_minus_sq(v.w);
        __builtin_nontemporal_store(r, &o4[i]);
    }

    // Scalar tail (n % 4 != 0); folds away for the 4096*8192 case.
    for (int j = n4 * 4 + tid; j < n; j += stride) {
        float v = __builtin_nontemporal_load(&x[j]);
        __builtin_nontemporal_store(neg_log_one_minus_sq(v), &out[j]);
    }
}

extern "C" void kernel_launch(void* const* d_in, const int* in_sizes, int n_in,
                              void* d_out, int out_size, void* d_ws, size_t ws_size,
                              hipStream_t stream) {
    const float* x = (const float*)d_in[0];
    float* out     = (float*)d_out;
    const int n    = in_sizes[0];        // 33,554,432 fp32 elements
    const int n4   = n >> 2;             // float4 count

    const int block = 256;               // 8 wave32s per workgroup
    // ~4 vec4 iterations per thread: enough blocks to fill every WGP many
    // times over while amortizing loop overhead.
    int grid = (n4 + block * 4 - 1) / (block * 4);
    if (grid < 1) grid = 1;

    GeneratorLoss_elemwise_kernel<<<grid, block, 0, stream>>>(x, out, n4, n);
}